// MultiBoxLoss_5669356831965
// MI455X (gfx1250) — compile-verified
//
#include <hip/hip_runtime.h>
#include <hip/hip_bf16.h>

#define BATCH   128
#define PRIORS  8732
#define NGT     16
#define NCLS    21
#define LOCD    10
#define TILE    384                       // priors per conf tile
#define NTILES  ((PRIORS + TILE - 1) / TILE)   // 23 (22 full + 284)
#define SEGS    4                         // tile-strided segments per batch -> 512 blocks

typedef __attribute__((ext_vector_type(4))) unsigned int u32x4;
typedef __attribute__((ext_vector_type(8))) int          i32x8;
typedef __attribute__((ext_vector_type(4))) int          i32x4;

// ---------------------------------------------------------------- conf lse via double-buffered TDM
// First function in file so the disasm snippet shows the tensor_load_to_lds path.
__device__ __forceinline__ void tdm_issue(const float* gsrc, unsigned ldsByte, int nElems) {
    unsigned long long ga = (unsigned long long)(uintptr_t)gsrc;
    u32x4 g0; i32x8 g1; i32x4 gz4; i32x8 gz8;
    g0[0] = 1u;                                                // count=1 valid D#
    g0[1] = ldsByte;                                           // lds_addr (bytes)
    g0[2] = (unsigned)(ga & 0xffffffffu);                      // global_addr[31:0]
    g0[3] = (unsigned)((ga >> 32) & 0x01ffffffu) | (2u << 30); // global_addr[56:32], type=2
    unsigned td0 = (unsigned)nElems;                           // 1-D run of f32
    g1[0] = (int)(2u << 16);                                   // data_size = 4 bytes
    g1[1] = (int)((td0 & 0xffffu) << 16);                      // tensor_dim0[15:0]
    g1[2] = (int)(((td0 >> 16) & 0xffffu) | (1u << 16));       // tensor_dim0[31:16], tensor_dim1=1
    g1[3] = (int)((td0 & 0xffffu) << 16);                      // tile_dim0 = n  (<= 8064, fits 16b)
    g1[4] = 1;                                                 // tile_dim1=1, tile_dim2=0
    g1[5] = (int)td0;                                          // tensor_dim0_stride
    g1[6] = 0; g1[7] = 0;
    gz4[0] = gz4[1] = gz4[2] = gz4[3] = 0;
    gz8[0] = gz8[1] = gz8[2] = gz8[3] = 0;
    gz8[4] = gz8[5] = gz8[6] = gz8[7] = 0;
    __builtin_amdgcn_tensor_load_to_lds(g0, g1, gz4, gz4, gz8, 0);   // TENSORcnt++
}

__global__ __launch_bounds__(256) void k_conf(const float* __restrict__ conf_data,
                                              const unsigned char* __restrict__ confT,
                                              float* __restrict__ lc,
                                              float* __restrict__ gLossC)
{
    __shared__ float tile[2][TILE * NCLS];    // 64512 B double buffer
    __shared__ float s_c;
    const int b = blockIdx.x / SEGS;
    const int s = blockIdx.x % SEGS;
    if (threadIdx.x == 0) s_c = 0.0f;

    const float* confb = conf_data + (size_t)b * PRIORS * NCLS;
    const unsigned lds0 = (unsigned)(uintptr_t)(&tile[0][0]);  // low 32b of flat = LDS byte offset
    const unsigned lds1 = (unsigned)(uintptr_t)(&tile[1][0]);

    // prologue: wave 0 kicks off DMA of the first tile of this segment
    if (threadIdx.x < 32 && s < NTILES) {
        int p0 = s * TILE;
        int cnt = min(TILE, PRIORS - p0);
        tdm_issue(confb + (size_t)p0 * NCLS, lds0, cnt * NCLS);
    }

    float csum = 0.0f;
    int buf = 0;
    for (int t = s; t < NTILES; t += SEGS) {
        const int p0 = t * TILE;
        const int count = min(TILE, PRIORS - p0);
        const int tn = t + SEGS;
        if (threadIdx.x < 32) {           // overlap: issue next DMA, retire current (in-order TDM)
            if (tn < NTILES) {
                int np0 = tn * TILE;
                int ncnt = min(TILE, PRIORS - np0);
                tdm_issue(confb + (size_t)np0 * NCLS, buf ? lds0 : lds1, ncnt * NCLS);
                __builtin_amdgcn_s_wait_tensorcnt(1);   // oldest transfer (this tile) done
            } else {
                __builtin_amdgcn_s_wait_tensorcnt(0);
            }
        }
        __syncthreads();                  // tile[buf] valid for all waves

        for (int r = threadIdx.x; r < count; r += 256) {
            const float* row = &tile[buf][r * NCLS];
            float m = row[0];
            #pragma unroll
            for (int c = 1; c < NCLS; ++c) m = fmaxf(m, row[c]);
            float sm = 0.0f;
            #pragma unroll
            for (int c = 0; c < NCLS; ++c) sm += __expf(row[c] - m);
            float lse = m + __logf(sm);
            int p = p0 + r;
            int ct = confT[(size_t)b * PRIORS + p];
            float lossall = lse - row[ct];
            lc[(size_t)b * PRIORS + p] = (ct > 0) ? 0.0f : fmaxf(lossall, 0.0f);
            if (ct > 0) csum += lossall;
        }
        __syncthreads();                  // all readers done before buf is re-filled next iter
        buf ^= 1;
    }
    atomicAdd(&s_c, csum);
    __syncthreads();
    if (threadIdx.x == 0) atomicAdd(gLossC, s_c);
}

// ---------------------------------------------------------------- init
__global__ void k_init(float* W, int* nposB) {
    int i = threadIdx.x + blockIdx.x * blockDim.x;
    if (i < 4) W[i] = 0.0f;
    if (i < BATCH) nposB[i] = 0;
}

// ---------------------------------------------------------------- match + loc loss
__global__ __launch_bounds__(256) void k_match(const float* __restrict__ loc_data,
                                               const float* __restrict__ priors,
                                               const float* __restrict__ targets,
                                               float* __restrict__ gLossL,
                                               float* __restrict__ gN,
                                               int*   __restrict__ nposB,
                                               unsigned char* __restrict__ confT)
{
    const int b   = blockIdx.x;
    const int tid = threadIdx.x;

    __shared__ float tgt[NGT][11];
    __shared__ float areaA[NGT];
    __shared__ unsigned long long gbest[NGT];   // (iou_bits<<32)|~p : max -> best iou, ties -> smallest p
    __shared__ float bto[PRIORS];
    __shared__ unsigned char bti[PRIORS];
    __shared__ float s_l;
    __shared__ unsigned s_np;

    if (tid < NGT * 11) ((float*)tgt)[tid] = targets[(size_t)b * NGT * 11 + tid];
    if (tid < NGT) gbest[tid] = 0ull;
    if (tid == 0) { s_l = 0.0f; s_np = 0u; }
    __syncthreads();
    if (tid < NGT)
        areaA[tid] = (tgt[tid][2] - tgt[tid][0]) * (tgt[tid][3] - tgt[tid][1]);
    __syncthreads();

    unsigned long long lmax[NGT];
    #pragma unroll
    for (int g = 0; g < NGT; ++g) lmax[g] = 0ull;

    for (int p = tid; p < PRIORS; p += 256) {
        __builtin_prefetch(priors + 4 * (p + 256), 0, 0);     // global_prefetch_b8
        float4 pr = *(const float4*)(priors + 4 * p);
        float px1 = pr.x - pr.z * 0.5f, py1 = pr.y - pr.w * 0.5f;
        float px2 = pr.x + pr.z * 0.5f, py2 = pr.y + pr.w * 0.5f;
        float areaB = pr.z * pr.w;
        float best = -1.0f; int bidx = 0;
        #pragma unroll
        for (int g = 0; g < NGT; ++g) {
            float ix = fminf(tgt[g][2], px2) - fmaxf(tgt[g][0], px1);
            float iy = fminf(tgt[g][3], py2) - fmaxf(tgt[g][1], py1);
            float inter = fmaxf(ix, 0.0f) * fmaxf(iy, 0.0f);
            float iou = inter / (areaA[g] + areaB - inter);
            if (iou > best) { best = iou; bidx = g; }          // first-index argmax (axis=0)
            unsigned long long pk =
                ((unsigned long long)__float_as_uint(iou) << 32) | (unsigned)(~p);
            lmax[g] = pk > lmax[g] ? pk : lmax[g];
        }
        bto[p] = best;
        bti[p] = (unsigned char)bidx;
    }
    #pragma unroll
    for (int g = 0; g < NGT; ++g) atomicMax(&gbest[g], lmax[g]);   // ds_max_u64
    __syncthreads();

    if (tid == 0) {   // sequential override: later j wins (matches reference loop)
        for (int j = 0; j < NGT; ++j) {
            int idx = (int)(~(unsigned)(gbest[j] & 0xffffffffull));
            bto[idx] = 2.0f;
            bti[idx] = (unsigned char)j;
        }
    }
    __syncthreads();

    float lsum = 0.0f; unsigned npos = 0;
    const float* locb = loc_data + (size_t)b * PRIORS * LOCD;
    for (int p = tid; p < PRIORS; p += 256) {
        int g = bti[p];
        int ct = (bto[p] < 0.5f) ? 0 : ((int)tgt[g][10] + 1);
        confT[(size_t)b * PRIORS + p] = (unsigned char)ct;
        if (ct > 0) {
            ++npos;
            float4 pr = *(const float4*)(priors + 4 * p);
            float pcx = pr.x, pcy = pr.y, pw = pr.z, ph = pr.w;
            float m[10];
            #pragma unroll
            for (int i = 0; i < 10; ++i) m[i] = tgt[g][i];
            float lt[10];
            lt[0] = ((m[0] + m[2]) * 0.5f - pcx) / (0.1f * pw);
            lt[1] = ((m[1] + m[3]) * 0.5f - pcy) / (0.1f * ph);
            lt[2] = __logf((m[2] - m[0]) / pw) * 5.0f;
            lt[3] = __logf((m[3] - m[1]) / ph) * 5.0f;
            lt[4] = __logf(m[4] / pw + 0.1f) * 5.0f;
            lt[5] = __logf(m[5] / ph + 0.1f) * 5.0f;
            lt[6] = __logf(m[6] / pw + 0.1f) * 5.0f;
            lt[7] = __logf(m[7] / ph + 0.1f) * 5.0f;
            lt[8] = (m[8] - pcx) / (0.1f * pw);
            lt[9] = (m[9] - pcy) / (0.1f * ph);
            const float* lp = locb + (size_t)p * LOCD;
            #pragma unroll
            for (int i = 0; i < 10; i += 2) {
                float2 v = *(const float2*)(lp + i);   // 8B aligned: 40p + 4i, i even
                float d0 = v.x - lt[i], d1 = v.y - lt[i + 1];
                float a0 = fabsf(d0), a1 = fabsf(d1);
                lsum += (a0 < 1.0f ? 0.5f * d0 * d0 : a0 - 0.5f);
                lsum += (a1 < 1.0f ? 0.5f * d1 * d1 : a1 - 0.5f);
            }
        }
    }
    atomicAdd(&s_l, lsum);
    atomicAdd(&s_np, npos);
    __syncthreads();
    if (tid == 0) {
        atomicAdd(gLossL, s_l);
        atomicAdd(gN, (float)s_np);
        nposB[b] = (int)s_np;
    }
}

// ---------------------------------------------------------------- hard-negative top-k (radix select)
__global__ __launch_bounds__(256) void k_select(const float* __restrict__ lc,
                                                const int* __restrict__ nposB,
                                                float* __restrict__ gLossC)
{
    const int b = blockIdx.x;
    const int tid = threadIdx.x;
    __shared__ unsigned hist[256];
    __shared__ unsigned sh_val, sh_mask;
    __shared__ int sh_k;
    __shared__ float s_sum;
    __shared__ unsigned s_cnt;

    int k = nposB[b] * 3;
    k = min(k, PRIORS - 1);
    if (k <= 0) return;                         // uniform per block

    const float* v = lc + (size_t)b * PRIORS;
    if (tid == 0) { sh_val = 0u; sh_mask = 0u; sh_k = k; s_sum = 0.0f; s_cnt = 0u; }

    for (int pass = 3; pass >= 0; --pass) {
        int shift = pass * 8;
        hist[tid] = 0u;
        __syncthreads();
        unsigned hv = sh_val, hm = sh_mask;
        for (int p = tid; p < PRIORS; p += 256) {
            unsigned bits = __float_as_uint(v[p]);      // lc >= 0 -> uint order == float order
            if ((bits & hm) == hv) atomicAdd(&hist[(bits >> shift) & 255u], 1u);
        }
        __syncthreads();
        if (tid == 0) {
            unsigned acc = 0; int krem = sh_k;
            for (int d = 255; d >= 0; --d) {
                if (acc + hist[d] >= (unsigned)krem) {
                    sh_k = krem - (int)acc;
                    sh_val |= ((unsigned)d) << shift;
                    break;
                }
                acc += hist[d];
            }
            sh_mask |= 255u << shift;
        }
        __syncthreads();
    }

    unsigned T = sh_val;
    float Tf = __uint_as_float(T);
    float ssum = 0.0f; unsigned scnt = 0;
    for (int p = tid; p < PRIORS; p += 256) {
        float x = v[p];
        if (__float_as_uint(x) > T) { ssum += x; ++scnt; }
    }
    atomicAdd(&s_sum, ssum);
    atomicAdd(&s_cnt, scnt);
    __syncthreads();
    if (tid == 0) {
        float add = s_sum + (float)(k - (int)s_cnt) * Tf;   // ties at T all contribute T
        atomicAdd(gLossC, add);
    }
}

// ---------------------------------------------------------------- finalize
__global__ void k_final(const float* __restrict__ W, float* __restrict__ out) {
    if (threadIdx.x == 0) {
        float N = W[2];
        out[0] = W[0] / N;
        out[1] = W[1] / N;
    }
}

// ---------------------------------------------------------------- launch
extern "C" void kernel_launch(void* const* d_in, const int* in_sizes, int n_in,
                              void* d_out, int out_size, void* d_ws, size_t ws_size,
                              hipStream_t stream) {
    const float* loc     = (const float*)d_in[0];   // (128, 8732, 10)
    const float* conf    = (const float*)d_in[1];   // (128, 8732, 21)
    const float* priors  = (const float*)d_in[2];   // (8732, 4)
    const float* targets = (const float*)d_in[3];   // (128, 16, 11)
    float* out = (float*)d_out;                     // 2 scalars

    // workspace layout: [0]=loss_l [1]=loss_c [2]=N [3]=pad | nposB[i32 x128] | confT[u8 BxP] | lc[f32 BxP]
    float* W = (float*)d_ws;
    int* nposB = (int*)d_ws + 4;
    unsigned char* confT = (unsigned char*)(nposB + BATCH);
    size_t lcoff = (16 + BATCH * 4 + (size_t)BATCH * PRIORS + 15) & ~(size_t)15;
    float* lc = (float*)((char*)d_ws + lcoff);

    k_init  <<<1, 256, 0, stream>>>(W, nposB);
    k_match <<<BATCH, 256, 0, stream>>>(loc, priors, targets, W + 0, W + 2, nposB, confT);
    k_conf  <<<BATCH * SEGS, 256, 0, stream>>>(conf, confT, lc, W + 1);
    k_select<<<BATCH, 256, 0, stream>>>(lc, nposB, W + 1);
    k_final <<<1, 32, 0, stream>>>(W, out);
}